// BIMPM_89412629168434
// MI455X (gfx1250) — compile-verified
//
#include <hip/hip_runtime.h>
#include <math.h>

#define Bn 32
#define Tn 192
#define En 100
#define Hn 100
#define Ln 20
#define Gn 400   // 4H
#define Dn 200   // 2H == 10L
#define BT (Bn*Tn)

typedef float v2f __attribute__((ext_vector_type(2)));
typedef float v8f __attribute__((ext_vector_type(8)));

// ---------------------------------------------------------------------------
// Generic batched fp32 WMMA GEMM: C[bt] = opA(A[bt]) * opB(B[bt / bdivB])
// Template-specialized on transA/transB. One wave computes a 16x32 tile
// (two accumulators sharing the A fragment), K stepped by 4 via
// V_WMMA_F32_16X16X4_F32.
//
// Preconditions (hold for every call site in this file):
//   M % 16 == 0, K % 4 == 0, lda/ldb even, base pointers 8B-aligned.
// Only N may be ragged: B columns are CLAMPED to N-1 (clamped columns
// produce garbage results that are never stored; columns of C are
// independent), so the K loop carries no predication at all.
//
// A frag layout: lanes 0-15 -> rows m with K = k0,k0+1 ; lanes 16-31 -> K+2
// C/D layout:    VGPR i -> row i (lanes 0-15) / row i+8 (lanes 16-31)
// ---------------------------------------------------------------------------
template <int TA, int TB>
__global__ __launch_bounds__(32)
void wmma_gemm_f32_t(const float* __restrict__ A, const float* __restrict__ B,
                     float* __restrict__ C,
                     int M, int N, int K, int lda, int ldb, int ldc,
                     long long sA, long long sB, long long sC, int bdivB)
{
    int bt = blockIdx.z;
    A += (long long)bt * sA;
    B += (long long)(bt / bdivB) * sB;
    C += (long long)bt * sC;

    const int tm    = blockIdx.x * 16;
    const int tn    = blockIdx.y * 32;
    const int lane  = threadIdx.x & 31;
    const int mloc  = lane & 15;
    const int khalf = (lane >> 4) * 2;   // 0 or 2

    const int ar = tm + mloc;            // always < M
    int bc0 = tn + mloc;      if (bc0 > N - 1) bc0 = N - 1;
    int bc1 = tn + 16 + mloc; if (bc1 > N - 1) bc1 = N - 1;

    const float* pa  = TA ? (A + (long long)khalf * lda + ar)
                          : (A + (long long)ar * lda + khalf);
    const float* pb0 = TB ? (B + (long long)bc0 * ldb + khalf)
                          : (B + (long long)khalf * ldb + bc0);
    const float* pb1 = TB ? (B + (long long)bc1 * ldb + khalf)
                          : (B + (long long)khalf * ldb + bc1);
    const long long stepA = TA ? (long long)4 * lda : 4;
    const long long stepB = TB ? 4 : (long long)4 * ldb;

    v8f acc0 = {}, acc1 = {};
#pragma unroll 2
    for (int k0 = 0; k0 < K; k0 += 4) {
        v2f a, b0, b1;
        if (TA) { a[0]  = pa[0];  a[1]  = pa[lda]; }
        else    { a = *(const v2f*)pa; }
        if (TB) { b0 = *(const v2f*)pb0; b1 = *(const v2f*)pb1; }
        else    { b0[0] = pb0[0]; b0[1] = pb0[ldb];
                  b1[0] = pb1[0]; b1[1] = pb1[ldb]; }
        acc0 = __builtin_amdgcn_wmma_f32_16x16x4_f32(
            false, a, false, b0, (short)0, acc0, false, false);
        acc1 = __builtin_amdgcn_wmma_f32_16x16x4_f32(
            false, a, false, b1, (short)0, acc1, false, false);
        pa += stepA; pb0 += stepB; pb1 += stepB;
    }

    const int half = lane >> 4, col = lane & 15;
    const int c0 = tn + col, c1 = tn + 16 + col;
#pragma unroll
    for (int i = 0; i < 8; ++i) {
        int r = tm + i + 8 * half;      // always < M
        if (c0 < N) C[(long long)r * ldc + c0] = acc0[i];
        if (c1 < N) C[(long long)r * ldc + c1] = acc1[i];
    }
}

// ---------------------------------------------------------------------------
// Embedding gather
// ---------------------------------------------------------------------------
__global__ void gather_emb(const float* __restrict__ emb,
                           const int* __restrict__ ids, float* __restrict__ xe)
{
    int idx = blockIdx.x * blockDim.x + threadIdx.x;
    if (idx >= BT * En) return;
    int e  = idx % En;
    int bt = idx / En;
    xe[idx] = emb[(long long)ids[bt] * En + e];
}

// Highway elementwise: out = sig(pb+bt)*relu(pa+bh) + (1-sig)*x
__global__ void highway_ew(const float* __restrict__ pa,
                           const float* __restrict__ pb,
                           const float* __restrict__ xe,
                           const float* __restrict__ bh,
                           const float* __restrict__ btv,
                           float* __restrict__ out)
{
    int idx = blockIdx.x * blockDim.x + threadIdx.x;
    if (idx >= BT * Hn) return;
    int j = idx % Hn;
    float tg = 1.f / (1.f + __expf(-(pb[idx] + btv[j])));
    float hr = fmaxf(pa[idx] + bh[j], 0.f);
    out[idx] = tg * hr + (1.f - tg) * xe[idx];
}

// ---------------------------------------------------------------------------
// LSTM recurrence: 4 independent scans (seq x dir) run as 4 blocks.
// h/c state in LDS (ping-pong), whh read from L2 (160KB, hot).
// ---------------------------------------------------------------------------
struct LstmJob {
    const float* xp;   // (B,T,4H) input projection (no biases)
    const float* whh;  // (4H,H)
    const float* bih;  // (4H)
    const float* bhh;  // (4H)
    float*       out;  // (B,T,2H)
    int out_off;       // 0 (fw) or H (bw)
    int rev;           // 1 = process reversed timeline
};
struct LstmJobs { LstmJob j[4]; };

__global__ __launch_bounds__(1024)
void lstm_scan4(LstmJobs jobs)
{
    LstmJob jb = jobs.j[blockIdx.x];
    __shared__ float hbuf[2][Bn * Hn];
    __shared__ float cbuf[Bn * Hn];
    int tid = threadIdx.x;
    for (int i = tid; i < Bn * Hn; i += blockDim.x) {
        hbuf[0][i] = 0.f; cbuf[i] = 0.f;
    }
    __syncthreads();
    int p = 0;
    for (int step = 0; step < Tn; ++step) {
        int t = jb.rev ? (Tn - 1 - step) : step;
        for (int idx = tid; idx < Bn * Hn; idx += blockDim.x) {
            int b = idx / Hn, hh = idx % Hn;
            const float* xpp = jb.xp + ((long long)b * Tn + t) * Gn;
            float gi = xpp[hh]          + jb.bih[hh]          + jb.bhh[hh];
            float gf = xpp[Hn + hh]     + jb.bih[Hn + hh]     + jb.bhh[Hn + hh];
            float gg = xpp[2 * Hn + hh] + jb.bih[2 * Hn + hh] + jb.bhh[2 * Hn + hh];
            float go = xpp[3 * Hn + hh] + jb.bih[3 * Hn + hh] + jb.bhh[3 * Hn + hh];
            const float* hrow = &hbuf[p][b * Hn];
            const float* wi = jb.whh + (long long)hh * Hn;
            const float* wf = jb.whh + (long long)(Hn + hh) * Hn;
            const float* wg = jb.whh + (long long)(2 * Hn + hh) * Hn;
            const float* wo = jb.whh + (long long)(3 * Hn + hh) * Hn;
            for (int k = 0; k < Hn; ++k) {
                float hv = hrow[k];
                gi += hv * wi[k]; gf += hv * wf[k];
                gg += hv * wg[k]; go += hv * wo[k];
            }
            float si = 1.f / (1.f + __expf(-gi));
            float sf = 1.f / (1.f + __expf(-gf));
            float so = 1.f / (1.f + __expf(-go));
            float tg = tanhf(gg);
            float cv = sf * cbuf[idx] + si * tg;
            float hv = so * tanhf(cv);
            cbuf[idx] = cv;
            hbuf[1 - p][idx] = hv;
            jb.out[((long long)b * Tn + t) * Dn + jb.out_off + hh] = hv;
        }
        __syncthreads();
        p = 1 - p;
    }
}

// ---------------------------------------------------------------------------
// Attention weight kernels
// ---------------------------------------------------------------------------
// per (b,t) row: w1 = softmax over s ; w1m = rowmax mask / count
__global__ void softmax_row(const float* __restrict__ attn,
                            float* __restrict__ w1, float* __restrict__ w1m)
{
    int idx = blockIdx.x * blockDim.x + threadIdx.x;
    if (idx >= BT) return;
    const float* row = attn + (long long)idx * Tn;
    float mx = -1e30f;
    for (int s = 0; s < Tn; ++s) mx = fmaxf(mx, row[s]);
    float sum = 0.f, cnt = 0.f;
    for (int s = 0; s < Tn; ++s) {
        sum += __expf(row[s] - mx);
        cnt += (row[s] >= mx) ? 1.f : 0.f;
    }
    float inv = 1.f / sum, invc = 1.f / cnt;
    for (int s = 0; s < Tn; ++s) {
        w1 [(long long)idx * Tn + s] = __expf(row[s] - mx) * inv;
        w1m[(long long)idx * Tn + s] = (row[s] >= mx) ? invc : 0.f;
    }
}

// per (b,s) column: w2 = softmax over t ; m2 = colmax mask (0/1)
__global__ void softmax_col(const float* __restrict__ attn,
                            float* __restrict__ w2, float* __restrict__ m2)
{
    int idx = blockIdx.x * blockDim.x + threadIdx.x;
    if (idx >= BT) return;
    int b = idx / Tn, s = idx % Tn;
    const float* base = attn + (long long)b * Tn * Tn + s;
    float mx = -1e30f;
    for (int t = 0; t < Tn; ++t) mx = fmaxf(mx, base[(long long)t * Tn]);
    float sum = 0.f;
    for (int t = 0; t < Tn; ++t) sum += __expf(base[(long long)t * Tn] - mx);
    float inv = 1.f / sum;
    for (int t = 0; t < Tn; ++t) {
        float v = base[(long long)t * Tn];
        long long o = ((long long)b * Tn + t) * Tn + s;
        w2[o] = __expf(v - mx) * inv;
        m2[o] = (v >= mx) ? 1.f : 0.f;
    }
}

// in-place: m2 -> m2 / m2.sum(axis=2)  (division kept even for 0-sum rows,
// matching the reference's 0/0 -> NaN semantics)
__global__ void row_normalize(float* __restrict__ m2)
{
    int idx = blockIdx.x * blockDim.x + threadIdx.x;
    if (idx >= BT) return;
    float* row = m2 + (long long)idx * Tn;
    float sum = 0.f;
    for (int s = 0; s < Tn; ++s) sum += row[s];
    float inv = 1.f / sum;
    for (int s = 0; s < Tn; ++s) row[s] *= inv;
}

// ---------------------------------------------------------------------------
// Multi-perspective cosine match (full / attentive / max-attentive)
// out[b,t,off+l] = cos(w[l]*v1[b,t], w[l]*v2[b,t2])
// ---------------------------------------------------------------------------
__global__ void match_cos(const float* __restrict__ v1, int v1off,
                          const float* __restrict__ v2, int v2off, int v2fix,
                          const float* __restrict__ w,
                          float* __restrict__ out, int outoff)
{
    int idx = blockIdx.x * blockDim.x + threadIdx.x;
    if (idx >= BT * Ln) return;
    int l = idx % Ln; int bt = idx / Ln;
    int b = bt / Tn, t = bt % Tn;
    int t2 = (v2fix >= 0) ? v2fix : t;
    const float* p1 = v1 + ((long long)b * Tn + t ) * Dn + v1off;
    const float* p2 = v2 + ((long long)b * Tn + t2) * Dn + v2off;
    const float* wl = w + l * Hn;
    float dot = 0.f, na = 0.f, nb = 0.f;
    for (int h = 0; h < Hn; ++h) {
        float a = wl[h] * p1[h];
        float c = wl[h] * p2[h];
        dot += a * c; na += a * a; nb += c * c;
    }
    float den = fmaxf(sqrtf(na) * sqrtf(nb), 1e-8f);
    out[((long long)b * Tn + t) * Dn + outoff + l] = dot / den;
}

// norms: out[b,t,l] = ||w[l]*v[b,t]||
__global__ void norm_wl(const float* __restrict__ v, int voff,
                        const float* __restrict__ w, float* __restrict__ out)
{
    int idx = blockIdx.x * blockDim.x + threadIdx.x;
    if (idx >= BT * Ln) return;
    int l = idx % Ln; int bt = idx / Ln;
    int b = bt / Tn, t = bt % Tn;
    const float* p = v + ((long long)b * Tn + t) * Dn + voff;
    const float* wl = w + l * Hn;
    float s = 0.f;
    for (int h = 0; h < Hn; ++h) { float a = wl[h] * p[h]; s += a * a; }
    out[((long long)b * Tn + t) * Ln + l] = sqrtf(s);
}

// As[b,l,t,h] = w[l,h]^2 * v[b,t,voff+h]   (fold both w factors into A)
__global__ void scale_a(const float* __restrict__ v, int voff,
                        const float* __restrict__ w, float* __restrict__ As)
{
    long long idx = (long long)blockIdx.x * blockDim.x + threadIdx.x;
    if (idx >= (long long)Bn * Ln * Tn * Hn) return;
    int h = (int)(idx % Hn); long long r = idx / Hn;
    int t = (int)(r % Tn); r /= Tn;
    int l = (int)(r % Ln); int b = (int)(r / Ln);
    float wv = w[l * Hn + h];
    As[idx] = wv * wv * v[((long long)b * Tn + t) * Dn + voff + h];
}

// reduce mm over t: out[b,s,off+l] (max + mean)  tmp = mm/(n1*n2+eps)
__global__ void mm_reduce_t(const float* __restrict__ mm,
                            const float* __restrict__ n1,
                            const float* __restrict__ n2,
                            float* __restrict__ out, int offmax, int offmean)
{
    int idx = blockIdx.x * blockDim.x + threadIdx.x;
    if (idx >= BT * Ln) return;
    int l = idx % Ln; int bs = idx / Ln;
    int b = bs / Tn, s = bs % Tn;
    float n2v = n2[((long long)b * Tn + s) * Ln + l];
    float mx = -1e30f, sum = 0.f;
    for (int t = 0; t < Tn; ++t) {
        float n1v = n1[((long long)b * Tn + t) * Ln + l];
        float v = mm[(((long long)(b * Ln + l)) * Tn + t) * Tn + s]
                  / (n1v * n2v + 1e-8f);
        mx = fmaxf(mx, v); sum += v;
    }
    float* o = out + ((long long)b * Tn + s) * Dn;
    o[offmax + l] = mx;
    o[offmean + l] = sum * (1.f / Tn);
}

// reduce mm over s: out[b,t,off+l] (max + mean)
__global__ void mm_reduce_s(const float* __restrict__ mm,
                            const float* __restrict__ n1,
                            const float* __restrict__ n2,
                            float* __restrict__ out, int offmax, int offmean)
{
    int idx = blockIdx.x * blockDim.x + threadIdx.x;
    if (idx >= BT * Ln) return;
    int l = idx % Ln; int btI = idx / Ln;
    int b = btI / Tn, t = btI % Tn;
    float n1v = n1[((long long)b * Tn + t) * Ln + l];
    const float* row = mm + (((long long)(b * Ln + l)) * Tn + t) * Tn;
    float mx = -1e30f, sum = 0.f;
    for (int s = 0; s < Tn; ++s) {
        float n2v = n2[((long long)b * Tn + s) * Ln + l];
        float v = row[s] / (n1v * n2v + 1e-8f);
        mx = fmaxf(mx, v); sum += v;
    }
    float* o = out + ((long long)b * Tn + t) * Dn;
    o[offmax + l] = mx;
    o[offmean + l] = sum * (1.f / Tn);
}

// ---------------------------------------------------------------------------
// Aggregate + FC head
// ---------------------------------------------------------------------------
__global__ void aggregate(const float* __restrict__ x1cc,
                          const float* __restrict__ x2cc,
                          float* __restrict__ feat)
{
    int idx = blockIdx.x * blockDim.x + threadIdx.x;
    if (idx >= Bn * 800) return;
    int b = idx / 800, j = idx % 800;
    int sec = j / Dn, c = j % Dn;
    const float* src = (sec < 2) ? x1cc : x2cc;
    bool ismax = (sec % 2) == 0;
    float mx = -1e30f, sum = 0.f;
    for (int t = 0; t < Tn; ++t) {
        float v = src[((long long)b * Tn + t) * Dn + c];
        mx = fmaxf(mx, v); sum += v;
    }
    feat[idx] = ismax ? mx : sum * (1.f / Tn);
}

__global__ void fc1_relu(const float* __restrict__ feat,
                         const float* __restrict__ w,
                         const float* __restrict__ bias,
                         float* __restrict__ out)
{
    int idx = blockIdx.x * blockDim.x + threadIdx.x;
    if (idx >= Bn * Hn) return;
    int b = idx / Hn, j = idx % Hn;
    float acc = bias[j];
    for (int k = 0; k < 8 * Hn; ++k)
        acc += feat[b * 8 * Hn + k] * w[k * Hn + j];
    out[idx] = fmaxf(acc, 0.f);
}

__global__ void out_softmax(const float* __restrict__ h,
                            const float* __restrict__ w,
                            const float* __restrict__ bias,
                            float* __restrict__ out)
{
    int b = blockIdx.x * blockDim.x + threadIdx.x;
    if (b >= Bn) return;
    float lg[4];
    for (int j = 0; j < 4; ++j) {
        float a = bias[j];
        for (int k = 0; k < Hn; ++k) a += h[b * Hn + k] * w[k * 4 + j];
        lg[j] = a;
    }
    float mx = fmaxf(fmaxf(lg[0], lg[1]), fmaxf(lg[2], lg[3]));
    float sum = 0.f;
    for (int j = 0; j < 4; ++j) { lg[j] = __expf(lg[j] - mx); sum += lg[j]; }
    float inv = 1.f / sum;
    for (int j = 0; j < 4; ++j) out[b * 4 + j] = lg[j] * inv;
}

// ---------------------------------------------------------------------------
// Host side
// ---------------------------------------------------------------------------
static inline void gemm(hipStream_t s, const float* A, const float* B, float* C,
                        int M, int N, int K, int lda, int ldb, int ldc,
                        long long sA, long long sB, long long sC,
                        int bdivB, int tA, int tB, int batch)
{
    dim3 g((unsigned)(M / 16), (unsigned)((N + 31) / 32), (unsigned)batch);
    if (!tA && !tB)
        wmma_gemm_f32_t<0, 0><<<g, dim3(32), 0, s>>>(A, B, C, M, N, K, lda, ldb, ldc, sA, sB, sC, bdivB);
    else if (!tA && tB)
        wmma_gemm_f32_t<0, 1><<<g, dim3(32), 0, s>>>(A, B, C, M, N, K, lda, ldb, ldc, sA, sB, sC, bdivB);
    else if (tA && !tB)
        wmma_gemm_f32_t<1, 0><<<g, dim3(32), 0, s>>>(A, B, C, M, N, K, lda, ldb, ldc, sA, sB, sC, bdivB);
    else
        wmma_gemm_f32_t<1, 1><<<g, dim3(32), 0, s>>>(A, B, C, M, N, K, lda, ldb, ldc, sA, sB, sC, bdivB);
}

extern "C" void kernel_launch(void* const* d_in, const int* in_sizes, int n_in,
                              void* d_out, int out_size, void* d_ws, size_t ws_size,
                              hipStream_t stream)
{
    (void)in_sizes; (void)n_in; (void)out_size; (void)ws_size;

    const float* emb         = (const float*)d_in[0];
    const float* hw_wh       = (const float*)d_in[1];
    const float* hw_bh       = (const float*)d_in[2];
    const float* hw_wt       = (const float*)d_in[3];
    const float* hw_bt       = (const float*)d_in[4];
    const float* enc_fw_wih  = (const float*)d_in[5];
    const float* enc_fw_whh  = (const float*)d_in[6];
    const float* enc_fw_bih  = (const float*)d_in[7];
    const float* enc_fw_bhh  = (const float*)d_in[8];
    const float* enc_bw_wih  = (const float*)d_in[9];
    const float* enc_bw_whh  = (const float*)d_in[10];
    const float* enc_bw_bih  = (const float*)d_in[11];
    const float* enc_bw_bhh  = (const float*)d_in[12];
    const float* comp_fw_wih = (const float*)d_in[13];
    const float* comp_fw_whh = (const float*)d_in[14];
    const float* comp_fw_bih = (const float*)d_in[15];
    const float* comp_fw_bhh = (const float*)d_in[16];
    const float* comp_bw_wih = (const float*)d_in[17];
    const float* comp_bw_whh = (const float*)d_in[18];
    const float* comp_bw_bih = (const float*)d_in[19];
    const float* comp_bw_bhh = (const float*)d_in[20];
    const float* match_w     = (const float*)d_in[21];
    const float* fc1_w       = (const float*)d_in[22];
    const float* fc1_b       = (const float*)d_in[23];
    const float* out_w       = (const float*)d_in[24];
    const float* out_b       = (const float*)d_in[25];
    const int*   x1_ids      = (const int*)d_in[26];
    const int*   x2_ids      = (const int*)d_in[27];
    float* outp = (float*)d_out;

    float* ws = (float*)d_ws;
    size_t off = 0;
    auto alloc = [&](size_t n) { float* p = ws + off; off += n; return p; };

    float* xe   = alloc((size_t)BT * En);
    float* pa   = alloc((size_t)BT * Hn);
    float* pb   = alloc((size_t)BT * Hn);
    float* x1h  = alloc((size_t)BT * Hn);
    float* x2h  = alloc((size_t)BT * Hn);
    float* xpA  = alloc((size_t)BT * Gn);
    float* xpB  = alloc((size_t)BT * Gn);
    float* xpC  = alloc((size_t)BT * Gn);
    float* xpD  = alloc((size_t)BT * Gn);
    float* x1   = alloc((size_t)BT * Dn);
    float* x2   = alloc((size_t)BT * Dn);
    float* attn = alloc((size_t)BT * Tn);
    float* w1   = alloc((size_t)BT * Tn);
    float* w2   = alloc((size_t)BT * Tn);
    float* w1m  = alloc((size_t)BT * Tn);
    float* m2   = alloc((size_t)BT * Tn);      // becomes w2m in place
    float* x1a  = alloc((size_t)BT * Dn);
    float* x2a  = alloc((size_t)BT * Dn);
    float* x1ma = alloc((size_t)BT * Dn);
    float* x2ma = alloc((size_t)BT * Dn);
    float* x1c  = alloc((size_t)BT * Dn);
    float* x2c  = alloc((size_t)BT * Dn);
    float* n1   = alloc((size_t)BT * Ln);
    float* n2   = alloc((size_t)BT * Ln);
    float* n1b  = alloc((size_t)BT * Ln);
    float* n2b  = alloc((size_t)BT * Ln);
    float* As   = alloc((size_t)Bn * Ln * Tn * Hn);
    float* mm   = alloc((size_t)Bn * Ln * Tn * Tn);
    float* x1cc = alloc((size_t)BT * Dn);
    float* x2cc = alloc((size_t)BT * Dn);
    float* feat = alloc((size_t)Bn * 800);
    float* fch  = alloc((size_t)Bn * Hn);

    auto blks = [](long long n) { return (unsigned)((n + 255) / 256); };

    // ---- embedding + highway (both sequences) ----
    gather_emb<<<blks((long long)BT * En), 256, 0, stream>>>(emb, x1_ids, xe);
    gemm(stream, xe, hw_wh, pa, BT, Hn, En, En, Hn, Hn, 0, 0, 0, 1, 0, 0, 1);
    gemm(stream, xe, hw_wt, pb, BT, Hn, En, En, Hn, Hn, 0, 0, 0, 1, 0, 0, 1);
    highway_ew<<<blks((long long)BT * Hn), 256, 0, stream>>>(pa, pb, xe, hw_bh, hw_bt, x1h);

    gather_emb<<<blks((long long)BT * En), 256, 0, stream>>>(emb, x2_ids, xe);
    gemm(stream, xe, hw_wh, pa, BT, Hn, En, En, Hn, Hn, 0, 0, 0, 1, 0, 0, 1);
    gemm(stream, xe, hw_wt, pb, BT, Hn, En, En, Hn, Hn, 0, 0, 0, 1, 0, 0, 1);
    highway_ew<<<blks((long long)BT * Hn), 256, 0, stream>>>(pa, pb, xe, hw_bh, hw_bt, x2h);

    // ---- encoder BiLSTM: input projections (transB: xp = x @ wih^T) ----
    gemm(stream, x1h, enc_fw_wih, xpA, BT, Gn, En, En, En, Gn, 0, 0, 0, 1, 0, 1, 1);
    gemm(stream, x1h, enc_bw_wih, xpB, BT, Gn, En, En, En, Gn, 0, 0, 0, 1, 0, 1, 1);
    gemm(stream, x2h, enc_fw_wih, xpC, BT, Gn, En, En, En, Gn, 0, 0, 0, 1, 0, 1, 1);
    gemm(stream, x2h, enc_bw_wih, xpD, BT, Gn, En, En, En, Gn, 0, 0, 0, 1, 0, 1, 1);

    {
        LstmJobs jobs;
        jobs.j[0] = { xpA, enc_fw_whh, enc_fw_bih, enc_fw_bhh, x1, 0,  0 };
        jobs.j[1] = { xpB, enc_bw_whh, enc_bw_bih, enc_bw_bhh, x1, Hn, 1 };
        jobs.j[2] = { xpC, enc_fw_whh, enc_fw_bih, enc_fw_bhh, x2, 0,  0 };
        jobs.j[3] = { xpD, enc_bw_whh, enc_bw_bih, enc_bw_bhh, x2, Hn, 1 };
        lstm_scan4<<<4, 1024, 0, stream>>>(jobs);
    }

    // ---- attention: attn[b] = x1[b] * x2[b]^T  (batched over B) ----
    gemm(stream, x1, x2, attn, Tn, Tn, Dn, Dn, Dn, Tn,
         (long long)Tn * Dn, (long long)Tn * Dn, (long long)Tn * Tn, 1, 0, 1, Bn);

    softmax_row<<<blks(BT), 256, 0, stream>>>(attn, w1, w1m);
    softmax_col<<<blks(BT), 256, 0, stream>>>(attn, w2, m2);
    row_normalize<<<blks(BT), 256, 0, stream>>>(m2);   // m2 -> w2m

    // ---- attentive sums (batched over B) ----
    gemm(stream, w1,  x2, x1a,  Tn, Dn, Tn, Tn, Dn, Dn,
         (long long)Tn * Tn, (long long)Tn * Dn, (long long)Tn * Dn, 1, 0, 0, Bn);
    gemm(stream, w2,  x1, x2a,  Tn, Dn, Tn, Tn, Dn, Dn,
         (long long)Tn * Tn, (long long)Tn * Dn, (long long)Tn * Dn, 1, 1, 0, Bn);
    gemm(stream, w1m, x2, x1ma, Tn, Dn, Tn, Tn, Dn, Dn,
         (long long)Tn * Tn, (long long)Tn * Dn, (long long)Tn * Dn, 1, 0, 0, Bn);
    gemm(stream, m2,  x1, x2ma, Tn, Dn, Tn, Tn, Dn, Dn,
         (long long)Tn * Tn, (long long)Tn * Dn, (long long)Tn * Dn, 1, 1, 0, Bn);

    // ---- cosine matches (full / attentive / max-attentive) ----
    long long nML = (long long)BT * Ln;
    match_cos<<<blks(nML), 256, 0, stream>>>(x1, 0,  x2,   0,  Tn - 1, match_w, x1c, 0);
    match_cos<<<blks(nML), 256, 0, stream>>>(x1, Hn, x2,   Hn, Tn - 1, match_w, x1c, 20);
    match_cos<<<blks(nML), 256, 0, stream>>>(x1, 0,  x1a,  0,  -1,     match_w, x1c, 40);
    match_cos<<<blks(nML), 256, 0, stream>>>(x1, Hn, x1a,  Hn, -1,     match_w, x1c, 60);
    match_cos<<<blks(nML), 256, 0, stream>>>(x1, 0,  x1ma, 0,  -1,     match_w, x1c, 160);
    match_cos<<<blks(nML), 256, 0, stream>>>(x1, Hn, x1ma, Hn, -1,     match_w, x1c, 180);
    match_cos<<<blks(nML), 256, 0, stream>>>(x2, 0,  x1,   0,  Tn - 1, match_w, x2c, 0);
    match_cos<<<blks(nML), 256, 0, stream>>>(x2, Hn, x1,   Hn, Tn - 1, match_w, x2c, 20);
    match_cos<<<blks(nML), 256, 0, stream>>>(x2, 0,  x2a,  0,  -1,     match_w, x2c, 40);
    match_cos<<<blks(nML), 256, 0, stream>>>(x2, Hn, x2a,  Hn, -1,     match_w, x2c, 60);
    match_cos<<<blks(nML), 256, 0, stream>>>(x2, 0,  x2ma, 0,  -1,     match_w, x2c, 160);
    match_cos<<<blks(nML), 256, 0, stream>>>(x2, Hn, x2ma, Hn, -1,     match_w, x2c, 180);

    // ---- maxpool match: norms, scaled-A, batched GEMM (B*L), reductions ----
    norm_wl<<<blks(nML), 256, 0, stream>>>(x1, 0,  match_w, n1);
    norm_wl<<<blks(nML), 256, 0, stream>>>(x2, 0,  match_w, n2);
    norm_wl<<<blks(nML), 256, 0, stream>>>(x1, Hn, match_w, n1b);
    norm_wl<<<blks(nML), 256, 0, stream>>>(x2, Hn, match_w, n2b);

    long long nAs = (long long)Bn * Ln * Tn * Hn;
    // forward halves -> x1c (mp1@80, mp2@100, ap1@120, ap2@140)
    scale_a<<<blks(nAs), 256, 0, stream>>>(x1, 0, match_w, As);
    gemm(stream, As, x2, mm, Tn, Tn, Hn, Hn, Dn, Tn,
         (long long)Tn * Hn, (long long)Tn * Dn, (long long)Tn * Tn, Ln, 0, 1, Bn * Ln);
    mm_reduce_t<<<blks(nML), 256, 0, stream>>>(mm, n1, n2, x1c, 80, 120);
    mm_reduce_s<<<blks(nML), 256, 0, stream>>>(mm, n1, n2, x1c, 100, 140);
    // backward halves -> x2c (mp3@80, mp4@100, ap3@120, ap4@140)
    scale_a<<<blks(nAs), 256, 0, stream>>>(x1, Hn, match_w, As);
    gemm(stream, As, x2 + Hn, mm, Tn, Tn, Hn, Hn, Dn, Tn,
         (long long)Tn * Hn, (long long)Tn * Dn, (long long)Tn * Tn, Ln, 0, 1, Bn * Ln);
    mm_reduce_t<<<blks(nML), 256, 0, stream>>>(mm, n1b, n2b, x2c, 80, 120);
    mm_reduce_s<<<blks(nML), 256, 0, stream>>>(mm, n1b, n2b, x2c, 100, 140);

    // ---- composition BiLSTM ----
    gemm(stream, x1c, comp_fw_wih, xpA, BT, Gn, Dn, Dn, Dn, Gn, 0, 0, 0, 1, 0, 1, 1);
    gemm(stream, x1c, comp_bw_wih, xpB, BT, Gn, Dn, Dn, Dn, Gn, 0, 0, 0, 1, 0, 1, 1);
    gemm(stream, x2c, comp_fw_wih, xpC, BT, Gn, Dn, Dn, Dn, Gn, 0, 0, 0, 1, 0, 1, 1);
    gemm(stream, x2c, comp_bw_wih, xpD, BT, Gn, Dn, Dn, Dn, Gn, 0, 0, 0, 1, 0, 1, 1);
    {
        LstmJobs jobs;
        jobs.j[0] = { xpA, comp_fw_whh, comp_fw_bih, comp_fw_bhh, x1cc, 0,  0 };
        jobs.j[1] = { xpB, comp_bw_whh, comp_bw_bih, comp_bw_bhh, x1cc, Hn, 1 };
        jobs.j[2] = { xpC, comp_fw_whh, comp_fw_bih, comp_fw_bhh, x2cc, 0,  0 };
        jobs.j[3] = { xpD, comp_bw_whh, comp_bw_bih, comp_bw_bhh, x2cc, Hn, 1 };
        lstm_scan4<<<4, 1024, 0, stream>>>(jobs);
    }

    // ---- aggregate + head ----
    aggregate<<<blks((long long)Bn * 800), 256, 0, stream>>>(x1cc, x2cc, feat);
    fc1_relu<<<blks((long long)Bn * Hn), 256, 0, stream>>>(feat, fc1_w, fc1_b, fch);
    out_softmax<<<1, 32, 0, stream>>>(fch, out_w, out_b, outp);
}